// SAGE_36816459661701
// MI455X (gfx1250) — compile-verified
//
#include <hip/hip_runtime.h>
#include <math.h>

typedef __attribute__((ext_vector_type(2))) float v2f;
typedef __attribute__((ext_vector_type(8))) float v8f;

#define HID 128

// ---------------------------------------------------------------- utilities
__global__ void zero_f32(float* p, long n) {
  long i = (long)blockIdx.x * blockDim.x + threadIdx.x;
  long stride = (long)gridDim.x * blockDim.x;
  for (; i < n; i += stride) p[i] = 0.0f;
}

__global__ void deg_kernel(const int* __restrict__ dst, float* __restrict__ deg, int E) {
  int i = blockIdx.x * blockDim.x + threadIdx.x;
  int s = gridDim.x * blockDim.x;
  for (; i < E; i += s) atomicAdd(&deg[dst[i]], 1.0f);
}

__global__ void invdeg_kernel(const float* __restrict__ deg, float* __restrict__ invd, int N) {
  int i = blockIdx.x * blockDim.x + threadIdx.x;
  if (i < N) invd[i] = 1.0f / fmaxf(deg[i], 1.0f);
}

// ------------------------------------------------ edge scatter (mean numerator)
// one thread per (edge, float4-chunk) for max memory-level parallelism; src/dst
// index loads broadcast from L2; agg (<=51MB) is L2-resident so atomics resolve there
template <int F>
__global__ void scatter_kernel(const float* __restrict__ x, const int* __restrict__ src,
                               const int* __restrict__ dst, float* __restrict__ agg,
                               int E) {
  constexpr int CH = F / 4;  // float4 chunks per edge
  long idx = (long)blockIdx.x * blockDim.x + threadIdx.x;
  if (idx >= (long)E * CH) return;
  int e = (int)(idx / CH);
  int c = (int)(idx % CH);
  int s = src[e], d = dst[e];
  float4 v = ((const float4*)(x + (size_t)s * F))[c];
  float* ad = agg + (size_t)d * F + 4 * c;
  atomicAdd(&ad[0], v.x);
  atomicAdd(&ad[1], v.y);
  atomicAdd(&ad[2], v.z);
  atomicAdd(&ad[3], v.w);
}

// ---------------------------------------------------- SAGE layer GEMM (WMMA f32)
// h_out[n,:] = relu( inv[n]*(agg[n,:]@Wl) + bl + xin[n,:]@Wr )
// 256 threads = 8 waves; block owns exactly 16 nodes (N % 16 == 0);
// wave w owns output cols [16w, 16w+16). Tiles staged via async Global->LDS DMA.
template <int K>
__global__ __launch_bounds__(256) void sage_gemm(
    const float* __restrict__ xin, const float* __restrict__ agg,
    const float* __restrict__ invd,
    const float* __restrict__ Wl, const float* __restrict__ bl,
    const float* __restrict__ Wr, float* __restrict__ hout) {
  constexpr int STR = K + 4;  // gcd(K+4,64)=4 -> conflict-free column reads
  constexpr int CH = K / 4;   // float4 chunks per tile row
  __shared__ float smem[2 * 16 * STR];  // [agg tile][x tile]
  __shared__ float sinv[16];
  const int tid = threadIdx.x;
  const int nodeBase = blockIdx.x * 16;

  if (tid < 16) sinv[tid] = invd[nodeBase + tid];

  // async-DMA the two 16 x K tiles straight into LDS (no VGPR round-trip);
  // all LDS destinations are 16B aligned (STR*4 = 528 or 80 bytes per row)
  for (int idx = tid; idx < 2 * 16 * CH; idx += 256) {
    int tile = idx / (16 * CH);
    int rem = idx % (16 * CH);
    int row = rem / CH;
    int c4 = rem % CH;
    const float* gsrc = (tile ? xin : agg) + (size_t)(nodeBase + row) * K + 4 * c4;
    unsigned ldsa =
        (unsigned)(uintptr_t)(smem + tile * 16 * STR + row * STR + 4 * c4);
    asm volatile("global_load_async_to_lds_b128 %0, %1, off"
                 :: "v"(ldsa), "v"((unsigned long long)(uintptr_t)gsrc)
                 : "memory");
  }
  asm volatile("s_wait_asynccnt 0" ::: "memory");
  __syncthreads();

  const int lane = tid & 31;
  const int wave = tid >> 5;
  const int colBase = wave * 16;
  const int row16 = lane & 15;  // A row / B column / D column
  const int hi = lane >> 4;     // lane group: 0 -> K=0,1 ; 1 -> K=2,3
  const int koff = hi * 2;

  v8f acc0 = {0.f, 0.f, 0.f, 0.f, 0.f, 0.f, 0.f, 0.f};  // agg @ Wl (unscaled)
  v8f acc1 = {0.f, 0.f, 0.f, 0.f, 0.f, 0.f, 0.f, 0.f};  // x   @ Wr
  const float* A0 = smem;             // agg tile
  const float* A1 = smem + 16 * STR;  // x tile

  // interleave the two independent GEMM passes -> alternating independent WMMAs
#pragma unroll
  for (int k = 0; k < K; k += 4) {
    v2f a0 = *(const v2f*)(A0 + row16 * STR + k + koff);
    v2f a1 = *(const v2f*)(A1 + row16 * STR + k + koff);
    v2f b0, b1;
    b0.x = Wl[(k + koff) * HID + colBase + row16];
    b0.y = Wl[(k + koff + 1) * HID + colBase + row16];
    b1.x = Wr[(k + koff) * HID + colBase + row16];
    b1.y = Wr[(k + koff + 1) * HID + colBase + row16];
    acc0 = __builtin_amdgcn_wmma_f32_16x16x4_f32(false, a0, false, b0, (short)0, acc0,
                                                 false, false);
    acc1 = __builtin_amdgcn_wmma_f32_16x16x4_f32(false, a1, false, b1, (short)0, acc1,
                                                 false, false);
  }

  float bias = bl[colBase + row16];
  float* outp = hout + (size_t)(nodeBase + hi * 8) * HID + colBase + row16;
#pragma unroll
  for (int r = 0; r < 8; ++r) {
    // C/D layout: VGPR r -> rows r (lanes 0-15), r+8 (lanes 16-31)
    float v = fmaf(acc0[r], sinv[hi * 8 + r], acc1[r] + bias);
    outp[(size_t)r * HID] = fmaxf(v, 0.0f);
  }
}

// -------------------------------------------------------- global max+mean pool
// 128 threads/block: thread t owns column t; coalesced row sweep; bitwise
// atomicMax is order-correct because post-ReLU values are >= 0
__global__ __launch_bounds__(128) void pool_kernel(const float* __restrict__ h,
                                                   float* __restrict__ pmax,
                                                   float* __restrict__ psum,
                                                   int N, int rowsPerBlock) {
  int c = threadIdx.x;
  int r0 = blockIdx.x * rowsPerBlock;
  int r1 = min(N, r0 + rowsPerBlock);
  float m = 0.0f, s = 0.0f;
  for (int r = r0; r < r1; ++r) {
    float v = h[(size_t)r * HID + c];
    m = fmaxf(m, v);
    s += v;
  }
  atomicMax((unsigned int*)&pmax[c], __float_as_uint(m));
  atomicAdd(&psum[c], s);
}

// ------------------------------------------------------------------- MLP head
__global__ __launch_bounds__(128) void head_kernel(
    const float* __restrict__ pmax, const float* __restrict__ psum,
    const float* __restrict__ W1, const float* __restrict__ b1,
    const float* __restrict__ W2, const float* __restrict__ b2,
    const float* __restrict__ W3, const float* __restrict__ b3,
    float* __restrict__ out, int N) {
  __shared__ float g[256];
  __shared__ float t1[128];
  __shared__ float t2[64];
  __shared__ float logit[2];
  int t = threadIdx.x;
  // g = x1 + x2 + x3, each [max(128) | mean(128)]
  g[t] = pmax[t] + pmax[128 + t] + pmax[256 + t];
  g[128 + t] = (psum[t] + psum[128 + t] + psum[256 + t]) / (float)N;
  __syncthreads();
  {  // lin1: [256] -> [128], relu
    float a = b1[t];
    for (int k = 0; k < 256; ++k) a += g[k] * W1[k * 128 + t];
    t1[t] = fmaxf(a, 0.0f);
  }
  __syncthreads();
  if (t < 64) {  // lin2: [128] -> [64], relu
    float a = b2[t];
    for (int k = 0; k < 128; ++k) a += t1[k] * W2[k * 64 + t];
    t2[t] = fmaxf(a, 0.0f);
  }
  __syncthreads();
  if (t < 2) {  // lin3: [64] -> [2], sigmoid
    float a = b3[t];
    for (int k = 0; k < 64; ++k) a += t2[k] * W3[k * 2 + t];
    logit[t] = 1.0f / (1.0f + expf(-a));
  }
  __syncthreads();
  if (t == 0) {  // log_softmax over 2 logits
    float s0 = logit[0], s1 = logit[1];
    float m = fmaxf(s0, s1);
    float lse = m + logf(expf(s0 - m) + expf(s1 - m));
    out[0] = s0 - lse;
    out[1] = s1 - lse;
  }
}

// ------------------------------------------------------------------- launcher
extern "C" void kernel_launch(void* const* d_in, const int* in_sizes, int n_in,
                              void* d_out, int out_size, void* d_ws, size_t ws_size,
                              hipStream_t stream) {
  (void)n_in; (void)out_size; (void)ws_size;
  const float* x = (const float*)d_in[0];
  const int* src = (const int*)d_in[2];
  const int* dst = (const int*)d_in[3];
  const float* W1l = (const float*)d_in[5];
  const float* b1l = (const float*)d_in[6];
  const float* W1r = (const float*)d_in[7];
  const float* W2l = (const float*)d_in[8];
  const float* b2l = (const float*)d_in[9];
  const float* W2r = (const float*)d_in[10];
  const float* W3l = (const float*)d_in[11];
  const float* b3l = (const float*)d_in[12];
  const float* W3r = (const float*)d_in[13];
  const float* lin1_W = (const float*)d_in[14];
  const float* lin1_b = (const float*)d_in[15];
  const float* lin2_W = (const float*)d_in[16];
  const float* lin2_b = (const float*)d_in[17];
  const float* lin3_W = (const float*)d_in[18];
  const float* lin3_b = (const float*)d_in[19];

  const int N = in_sizes[0] / 16;  // 100000 (multiple of 16)
  const int E = in_sizes[1];       // 1600000

  // workspace layout (floats)
  float* ws = (float*)d_ws;
  size_t o = 0;
  float* agg  = ws + o; o += (size_t)N * HID;
  float* h1   = ws + o; o += (size_t)N * HID;
  float* h2   = ws + o; o += (size_t)N * HID;
  float* deg  = ws + o; o += N;
  float* invd = ws + o; o += N;
  float* pmax = ws + o; o += 3 * HID;
  float* psum = ws + o; o += 3 * HID;

  const int TB = 256;
  auto gs = [](long n) { return (int)((n + 255) / 256); };
  const int nodeBlocks = N / 16;  // 6250, exact
  const int poolRows = 256;
  const int poolBlocks = (N + poolRows - 1) / poolRows;

  // degrees (shared across layers)
  zero_f32<<<gs(N), TB, 0, stream>>>(deg, N);
  deg_kernel<<<gs(E), TB, 0, stream>>>(dst, deg, E);
  invdeg_kernel<<<gs(N), TB, 0, stream>>>(deg, invd, N);
  // pool accumulators (pmax||psum contiguous, zero-init; values are >= 0)
  zero_f32<<<gs(6 * HID), TB, 0, stream>>>(pmax, 6 * HID);

  // ---- layer 1 (F_IN=16 -> 128)
  zero_f32<<<gs((long)N * 16), TB, 0, stream>>>(agg, (long)N * 16);
  scatter_kernel<16><<<gs((long)E * 4), TB, 0, stream>>>(x, src, dst, agg, E);
  sage_gemm<16><<<nodeBlocks, 256, 0, stream>>>(x, agg, invd, W1l, b1l, W1r, h1);
  pool_kernel<<<poolBlocks, 128, 0, stream>>>(h1, pmax + 0, psum + 0, N, poolRows);

  // ---- layer 2 (128 -> 128)
  zero_f32<<<gs((long)N * HID), TB, 0, stream>>>(agg, (long)N * HID);
  scatter_kernel<HID><<<gs((long)E * 32), TB, 0, stream>>>(h1, src, dst, agg, E);
  sage_gemm<HID><<<nodeBlocks, 256, 0, stream>>>(h1, agg, invd, W2l, b2l, W2r, h2);
  pool_kernel<<<poolBlocks, 128, 0, stream>>>(h2, pmax + HID, psum + HID, N, poolRows);

  // ---- layer 3 (128 -> 128), ping-pong back into h1
  zero_f32<<<gs((long)N * HID), TB, 0, stream>>>(agg, (long)N * HID);
  scatter_kernel<HID><<<gs((long)E * 32), TB, 0, stream>>>(h2, src, dst, agg, E);
  sage_gemm<HID><<<nodeBlocks, 256, 0, stream>>>(h2, agg, invd, W3l, b3l, W3r, h1);
  pool_kernel<<<poolBlocks, 128, 0, stream>>>(h1, pmax + 2 * HID, psum + 2 * HID, N, poolRows);

  // ---- head MLP
  head_kernel<<<1, 128, 0, stream>>>(pmax, psum, lin1_W, lin1_b, lin2_W, lin2_b,
                                     lin3_W, lin3_b, (float*)d_out, N);
}